// GATActorWithLaser_29188597743647
// MI455X (gfx1250) — compile-verified
//
#include <hip/hip_runtime.h>
#include <hip/hip_bf16.h>
#include <math.h>

typedef float v2f __attribute__((ext_vector_type(2)));
typedef float v8f __attribute__((ext_vector_type(8)));

#define NNODES 6144
#define NG     512
#define PP     12
#define AAG    4
#define NH     8
#define CH     64
#define DMODEL 512
#define NEDGE  67584
#define NLID   20

// ---------------------------------------------------------------------------
// Generic fp32 WMMA GEMM: C[M,Ncol] = A[M,K] @ B[K,Ncol] (+bias)(+relu)
// One wave computes a 16x64 tile using V_WMMA_F32_16X16X4_F32.
// M %16==0, Ncol %64==0, K %4==0. A rows 8B aligned (K even).
// ---------------------------------------------------------------------------
__global__ __launch_bounds__(256)
void k_gemm(const float* __restrict__ A, const float* __restrict__ B,
            const float* __restrict__ bias, float* __restrict__ C,
            int M, int K, int Ncol, int relu) {
  const int lane = threadIdx.x & 31;
  const int wave = blockIdx.x * (blockDim.x >> 5) + (threadIdx.x >> 5);
  const int colTiles = Ncol >> 6;
  const int rowTile = wave / colTiles;
  const int colTile = wave - rowTile * colTiles;
  if (rowTile * 16 >= M) return;               // wave-uniform exit
  const int l16 = lane & 15;
  const int kh  = lane >> 4;                   // 0 or 1 (K-half select)
  const float* Arow = A + (size_t)(rowTile * 16 + l16) * K;
  const float* Bcol = B + colTile * 64 + l16;
  v8f acc0 = {}, acc1 = {}, acc2 = {}, acc3 = {};
  for (int k0 = 0; k0 < K; k0 += 4) {
    const int ka = k0 + 2 * kh;                // K rows this lane supplies
    v2f a = *(const v2f*)(Arow + ka);          // A[m, ka], A[m, ka+1]
    const float* b0p = Bcol + (size_t)ka * Ncol;
    const float* b1p = b0p + Ncol;
    v2f b0 = { b0p[0],  b1p[0]  };
    v2f b1 = { b0p[16], b1p[16] };
    v2f b2 = { b0p[32], b1p[32] };
    v2f b3 = { b0p[48], b1p[48] };
    acc0 = __builtin_amdgcn_wmma_f32_16x16x4_f32(false, a, false, b0, (short)0, acc0, false, false);
    acc1 = __builtin_amdgcn_wmma_f32_16x16x4_f32(false, a, false, b1, (short)0, acc1, false, false);
    acc2 = __builtin_amdgcn_wmma_f32_16x16x4_f32(false, a, false, b2, (short)0, acc2, false, false);
    acc3 = __builtin_amdgcn_wmma_f32_16x16x4_f32(false, a, false, b3, (short)0, acc3, false, false);
  }
  // C/D layout: VGPR g -> M = g + 8*(lane>=16), N = lane&15
  const int mbase = rowTile * 16 + 8 * kh;
  float* Cp = C + colTile * 64 + l16;
  const float* bp = bias ? bias + colTile * 64 + l16 : nullptr;
#pragma unroll
  for (int g = 0; g < 8; ++g) {
    size_t roff = (size_t)(mbase + g) * Ncol;
    float v0 = acc0[g], v1 = acc1[g], v2 = acc2[g], v3 = acc3[g];
    if (bp) { v0 += bp[0]; v1 += bp[16]; v2 += bp[32]; v3 += bp[48]; }
    if (relu) {
      v0 = fmaxf(v0, 0.f); v1 = fmaxf(v1, 0.f);
      v2 = fmaxf(v2, 0.f); v3 = fmaxf(v3, 0.f);
    }
    Cp[roff] = v0; Cp[roff + 16] = v1; Cp[roff + 32] = v2; Cp[roff + 48] = v3;
  }
}

// ---------------------------------------------------------------------------
// Lidar CNN + embedding + reconstruction decoder; one wave per node.
// Writes xg0 (N,32): [non_lidar(6) | lid_emb(16) | last_act(3) | zeros(7)]
// and recon (N,20).
// ---------------------------------------------------------------------------
__global__ __launch_bounds__(256)
void k_lidar(const float* __restrict__ x,
             const float* __restrict__ c1w, const float* __restrict__ c1b,
             const float* __restrict__ c2w, const float* __restrict__ c2b,
             const float* __restrict__ lew, const float* __restrict__ leb,
             const float* __restrict__ d1w, const float* __restrict__ d1b,
             const float* __restrict__ d2w, const float* __restrict__ d2b,
             float* __restrict__ xg0, float* __restrict__ rc) {
  __shared__ float s_raw[8][24];
  __shared__ float s_pool[8][160];
  __shared__ float s_h[8][32];
  __shared__ float s_le[8][16];
  __shared__ float s_d1[8][64];
  const int wave = threadIdx.x >> 5, lane = threadIdx.x & 31;
  const int n = blockIdx.x * 8 + wave;
  const float* xr = x + (size_t)n * 29;
  if (lane < 24) s_raw[wave][lane] = (lane >= 2 && lane < 22) ? xr[lane - 2] : 0.f;
  __syncthreads();
  // conv1 (k=5,pad=2) + relu + maxpool2 fused -> 16ch x 10
  for (int idx = lane; idx < 160; idx += 32) {
    int ch = idx / 10, q = idx - ch * 10;
    float c0 = c1b[ch], c1 = c1b[ch];
#pragma unroll
    for (int t = 0; t < 5; ++t) {
      float w = c1w[ch * 5 + t];
      c0 += w * s_raw[wave][2 * q + t];
      c1 += w * s_raw[wave][2 * q + 1 + t];
    }
    s_pool[wave][idx] = fmaxf(fmaxf(c0, c1), 0.f);
  }
  __syncthreads();
  // conv2 (k=3,pad=1) + relu + mean over length -> 32
  {
    int oc = lane;
    float acc = 0.f;
    for (int pos = 0; pos < 10; ++pos) {
      float v = c2b[oc];
      for (int ic = 0; ic < 16; ++ic) {
        const float* pr = &s_pool[wave][ic * 10];
        const float* wr = c2w + oc * 48 + ic * 3;
        if (pos > 0) v += wr[0] * pr[pos - 1];
        v += wr[1] * pr[pos];
        if (pos < 9) v += wr[2] * pr[pos + 1];
      }
      acc += fmaxf(v, 0.f);
    }
    s_h[wave][oc] = acc * 0.1f;
  }
  __syncthreads();
  if (lane < 16) {                       // lid_emb = relu(h @ le_w + le_b)
    float v = leb[lane];
    for (int k = 0; k < 32; ++k) v += s_h[wave][k] * lew[k * 16 + lane];
    s_le[wave][lane] = fmaxf(v, 0.f);
  }
  __syncthreads();
  {                                      // dec1: 16 -> 64, relu
    int j0 = lane, j1 = lane + 32;
    float v0 = d1b[j0], v1 = d1b[j1];
    for (int k = 0; k < 16; ++k) {
      float lv = s_le[wave][k];
      v0 += lv * d1w[k * 64 + j0];
      v1 += lv * d1w[k * 64 + j1];
    }
    s_d1[wave][j0] = fmaxf(v0, 0.f);
    s_d1[wave][j1] = fmaxf(v1, 0.f);
  }
  __syncthreads();
  if (lane < NLID) {                     // dec2: 64 -> 20
    float v = d2b[lane];
    for (int k = 0; k < 64; ++k) v += s_d1[wave][k] * d2w[k * 20 + lane];
    rc[(size_t)n * NLID + lane] = v;
  }
  float xv;
  if (lane < 6)       xv = xr[20 + lane];
  else if (lane < 22) xv = s_le[wave][lane - 6];
  else if (lane < 25) xv = xr[26 + (lane - 22)];
  else                xv = 0.f;
  xg0[(size_t)n * 32 + lane] = xv;
}

// loop_attr[n] = mean over in-edges of edge_attr (fixed K_{P} minus self graph)
__global__ void k_loopattr(const float* __restrict__ ea, float* __restrict__ la) {
  int n = blockIdx.x * blockDim.x + threadIdx.x;
  if (n >= NNODES) return;
  int g = n / PP, j = n - g * PP;
  const float* eg = ea + (size_t)g * (PP * (PP - 1));
  float s = 0.f;
  for (int i = 0; i < PP; ++i) {
    if (i == j) continue;
    int col = j - (j > i ? 1 : 0);
    s += eg[i * (PP - 1) + col];
  }
  la[n] = s / (float)(PP - 1);
}

// kE[h] = sum_c We[h*64+c] * a_edge[h*64+c]
__global__ void k_ke(const float* __restrict__ We, const float* __restrict__ ae,
                     float* __restrict__ out) {
  int h = threadIdx.x;
  if (h < NH) {
    float s = 0.f;
    for (int c = 0; c < CH; ++c) s += We[h * CH + c] * ae[h * CH + c];
    out[h] = s;
  }
}

// zero-pad g1_W (25,512) -> (32,512)
__global__ void k_padw1(const float* __restrict__ w, float* __restrict__ wp) {
  int t = blockIdx.x * blockDim.x + threadIdx.x;
  if (t >= 32 * DMODEL) return;
  wp[t] = (t < 25 * DMODEL) ? w[t] : 0.f;
}

// asrc/adst per (node, head): 64-dot products against xp
__global__ void k_dots(const float* __restrict__ xp, const float* __restrict__ aw,
                       const float* __restrict__ dw, float* __restrict__ as_,
                       float* __restrict__ ad_) {
  int t = blockIdx.x * blockDim.x + threadIdx.x;
  if (t >= NNODES * NH) return;
  int n = t >> 3, h = t & 7;
  const float* xr = xp + (size_t)n * DMODEL + h * CH;
  const float* a = aw + h * CH;
  const float* d = dw + h * CH;
  float s = 0.f, q = 0.f;
  for (int c = 0; c < CH; ++c) { float v = xr[c]; s += v * a[c]; q += v * d[c]; }
  as_[t] = s; ad_[t] = q;
}

// Attention softmax + aggregation + bias + relu. One wave per (node, head).
__global__ __launch_bounds__(256)
void k_agg(const float* __restrict__ xp, const float* __restrict__ as_,
           const float* __restrict__ ad_, const float* __restrict__ ea,
           const float* __restrict__ la, const float* __restrict__ kE,
           const float* __restrict__ bias, float* __restrict__ xout) {
  int wid = blockIdx.x * 8 + (threadIdx.x >> 5);
  int lane = threadIdx.x & 31;
  int n = wid >> 3, h = wid & 7;
  int g = n / PP, j = n - g * PP;
  float adn = ad_[n * NH + h];
  float ke = kE[h];
  float alpha[PP];
  int sn[PP];
  const float* eg = ea + (size_t)g * (PP * (PP - 1));
#pragma unroll
  for (int t = 0; t < PP - 1; ++t) {           // 11 in-edges
    int i = t + (t >= j ? 1 : 0);
    int col = j - (j > i ? 1 : 0);
    float eav = eg[i * (PP - 1) + col];
    int s = g * PP + i;
    float al = as_[s * NH + h] + adn + eav * ke;
    alpha[t] = al > 0.f ? al : 0.2f * al;      // leaky_relu(0.2)
    sn[t] = s;
  }
  {                                            // self loop
    float al = as_[n * NH + h] + adn + la[n] * ke;
    alpha[PP - 1] = al > 0.f ? al : 0.2f * al;
    sn[PP - 1] = n;
  }
  float amax = alpha[0];
#pragma unroll
  for (int t = 1; t < PP; ++t) amax = fmaxf(amax, alpha[t]);
  float den = 0.f;
#pragma unroll
  for (int t = 0; t < PP; ++t) { alpha[t] = expf(alpha[t] - amax); den += alpha[t]; }
  float inv = 1.f / den;
  int c0 = lane, c1 = lane + 32;
  float o0 = bias[h * CH + c0], o1 = bias[h * CH + c1];
#pragma unroll
  for (int t = 0; t < PP; ++t) {
    const float* xr = xp + (size_t)sn[t] * DMODEL + h * CH;
    float cf = alpha[t] * inv;
    o0 += cf * xr[c0];
    o1 += cf * xr[c1];
  }
  float* yo = xout + (size_t)n * DMODEL + h * CH;
  yo[c0] = fmaxf(o0, 0.f);
  yo[c1] = fmaxf(o1, 0.f);
}

// glob[g,c] = mean over P nodes
__global__ void k_glob(const float* __restrict__ xe, float* __restrict__ gl) {
  int t = blockIdx.x * blockDim.x + threadIdx.x;
  if (t >= NG * DMODEL) return;
  int g = t >> 9, c = t & 511;
  const float* base = xe + (size_t)g * PP * DMODEL + c;
  float s = 0.f;
  for (int p = 0; p < PP; ++p) s += base[p * DMODEL];
  gl[t] = s * (1.f / (float)PP);
}

// comb[r, 0:512] = agent_emb, comb[r, 512:1024] = glob[g]
__global__ void k_comb(const float* __restrict__ xe, const float* __restrict__ gl,
                       float* __restrict__ cb) {
  int t = blockIdx.x * blockDim.x + threadIdx.x;
  if (t >= NG * AAG * 1024) return;
  int r = t >> 10, c = t & 1023;
  int g = r >> 2, a = r & 3;
  cb[t] = (c < DMODEL) ? xe[(size_t)(g * PP + a) * DMODEL + c]
                       : gl[(size_t)g * DMODEL + (c - DMODEL)];
}

// fc2 + tanh/sigmoid heads -> mean (out[0:6144]) and std (out[6144:12288])
__global__ void k_head(const float* __restrict__ hh, const float* __restrict__ w2,
                       const float* __restrict__ b2, float* __restrict__ out) {
  int r = blockIdx.x * blockDim.x + threadIdx.x;
  if (r >= NG * AAG) return;
  const float* hr = hh + (size_t)r * 256;
  float ms[6];
#pragma unroll
  for (int o = 0; o < 6; ++o) ms[o] = b2[o];
  for (int k = 0; k < 256; ++k) {
    float hv = hr[k];
    const float* wr = w2 + k * 6;
#pragma unroll
    for (int o = 0; o < 6; ++o) ms[o] += hv * wr[o];
  }
  const float lim0 = 1.f, lim1 = 1.f, lim2 = 3.14159f;
  out[r * 3 + 0] = tanhf(ms[0]) * lim0;
  out[r * 3 + 1] = tanhf(ms[1]) * lim1;
  out[r * 3 + 2] = tanhf(ms[2]) * lim2;
#pragma unroll
  for (int o = 0; o < 3; ++o) {
    float sg = 1.f / (1.f + expf(-ms[3 + o]));
    out[NG * AAG * 3 + r * 3 + o] = 0.01f + sg * (0.45f - 0.01f) + 1e-5f;
  }
}

// gather raw_lidar[agent_idx] and recon[agent_idx] into d_out tail
__global__ void k_gather(const float* __restrict__ x, const float* __restrict__ rc,
                         float* __restrict__ out) {
  int t = blockIdx.x * blockDim.x + threadIdx.x;
  if (t >= NG * AAG * NLID) return;
  int r = t / NLID, j = t - r * NLID;
  int g = r >> 2, a = r & 3;
  int node = g * PP + a;
  out[2 * NG * AAG * 3 + t] = x[(size_t)node * 29 + j];
  out[2 * NG * AAG * 3 + NG * AAG * NLID + t] = rc[(size_t)node * NLID + j];
}

extern "C" void kernel_launch(void* const* d_in, const int* in_sizes, int n_in,
                              void* d_out, int out_size, void* d_ws, size_t ws_size,
                              hipStream_t stream) {
  (void)in_sizes; (void)n_in; (void)out_size; (void)ws_size;
  const float* x      = (const float*)d_in[0];
  const float* eattr  = (const float*)d_in[1];
  const float* c1w    = (const float*)d_in[2];
  const float* c1b    = (const float*)d_in[3];
  const float* c2w    = (const float*)d_in[4];
  const float* c2b    = (const float*)d_in[5];
  const float* lew    = (const float*)d_in[6];
  const float* leb    = (const float*)d_in[7];
  const float* d1w    = (const float*)d_in[8];
  const float* d1b    = (const float*)d_in[9];
  const float* d2w    = (const float*)d_in[10];
  const float* d2b    = (const float*)d_in[11];
  const float* fc1w   = (const float*)d_in[12];
  const float* fc1b   = (const float*)d_in[13];
  const float* fc2w   = (const float*)d_in[14];
  const float* fc2b   = (const float*)d_in[15];
  const float* gW[3]    = {(const float*)d_in[16], (const float*)d_in[22], (const float*)d_in[28]};
  const float* gasrc[3] = {(const float*)d_in[17], (const float*)d_in[23], (const float*)d_in[29]};
  const float* gadst[3] = {(const float*)d_in[18], (const float*)d_in[24], (const float*)d_in[30]};
  const float* gWe[3]   = {(const float*)d_in[19], (const float*)d_in[25], (const float*)d_in[31]};
  const float* gae[3]   = {(const float*)d_in[20], (const float*)d_in[26], (const float*)d_in[32]};
  const float* gb[3]    = {(const float*)d_in[21], (const float*)d_in[27], (const float*)d_in[33]};
  float* out = (float*)d_out;
  float* ws  = (float*)d_ws;

  // workspace layout (floats)
  size_t oXA = 0;
  size_t oXB = oXA + (size_t)NNODES * DMODEL;
  size_t oXP = oXB + (size_t)NNODES * DMODEL;
  size_t oRC = oXP + (size_t)NNODES * DMODEL;
  size_t oLA = oRC + (size_t)NNODES * NLID;
  size_t oAS = oLA + NNODES;
  size_t oAD = oAS + (size_t)NNODES * NH;
  size_t oKE = oAD + (size_t)NNODES * NH;
  size_t oWP = oKE + 32;
  size_t oGB = oWP + 32 * DMODEL;
  size_t oCB = oGB + (size_t)NG * DMODEL;
  size_t oHHm = oCB + (size_t)NG * AAG * 1024;
  float *XA = ws + oXA, *XB = ws + oXB, *XP = ws + oXP, *RC = ws + oRC;
  float *LA = ws + oLA, *AS = ws + oAS, *AD = ws + oAD, *KE = ws + oKE;
  float *WP = ws + oWP, *GB = ws + oGB, *CB = ws + oCB, *HHB = ws + oHHm;

  // prep
  k_padw1<<<64, 256, 0, stream>>>(gW[0], WP);
  k_ke<<<1, 32, 0, stream>>>(gWe[0], gae[0], KE + 0);
  k_ke<<<1, 32, 0, stream>>>(gWe[1], gae[1], KE + 8);
  k_ke<<<1, 32, 0, stream>>>(gWe[2], gae[2], KE + 16);
  k_lidar<<<NNODES / 8, 256, 0, stream>>>(x, c1w, c1b, c2w, c2b, lew, leb,
                                          d1w, d1b, d2w, d2b, XA, RC);
  k_loopattr<<<NNODES / 256, 256, 0, stream>>>(eattr, LA);

  // GAT layers: (in, W, K) -> XP -> agg -> out
  const float* Xin[3] = {XA, XB, XA};
  float*       Xout[3] = {XB, XA, XB};
  const float* Wl[3] = {WP, gW[1], gW[2]};
  const int    Kl[3] = {32, DMODEL, DMODEL};
  for (int l = 0; l < 3; ++l) {
    int waves = (NNODES / 16) * (DMODEL / 64);          // 3072
    k_gemm<<<waves / 8, 256, 0, stream>>>(Xin[l], Wl[l], nullptr, XP,
                                          NNODES, Kl[l], DMODEL, 0);
    k_dots<<<NNODES * NH / 256, 256, 0, stream>>>(XP, gasrc[l], gadst[l], AS, AD);
    k_agg<<<NNODES * NH / 8, 256, 0, stream>>>(XP, AS, AD, eattr, LA, KE + 8 * l,
                                               gb[l], Xout[l]);
  }

  // pooling + policy head
  k_glob<<<NG * DMODEL / 256, 256, 0, stream>>>(XB, GB);
  k_comb<<<NG * AAG * 1024 / 256, 256, 0, stream>>>(XB, GB, CB);
  {
    int waves = (NG * AAG / 16) * (256 / 64);           // 512
    k_gemm<<<waves / 8, 256, 0, stream>>>(CB, fc1w, fc1b, HHB,
                                          NG * AAG, 1024, 256, 1);
  }
  k_head<<<NG * AAG / 256, 256, 0, stream>>>(HHB, fc2w, fc2b, out);
  k_gather<<<NG * AAG * NLID / 256, 256, 0, stream>>>(x, RC, out);
}